// A3TGCN2_79723182948866
// MI455X (gfx1250) — compile-verified
//
#include <hip/hip_runtime.h>
#include <math.h>

#define BATCH   4
#define N_NODES 50000
#define FT      24          // IN_CH * PERIODS = 2*12 contiguous floats per (b,node)
#define PERIODS 12

typedef __attribute__((ext_vector_type(2))) float v2f;
typedef __attribute__((ext_vector_type(8))) float v8f;

// ---------------------------------------------------------------------------
// Fuse W_g[2x32] @ lin_g_w[0:32,0:32] -> M_g[2x32]; fused bias; softmax(att).
// ws layout (floats): M[0..31]=Mz row0, [32..63]=Mz row1, [64..95]=cz,
//                     [96..127]=Mh row0, [128..159]=Mh row1, [160..191]=ch,
//                     [192..203]=probs
// ---------------------------------------------------------------------------
__global__ void prep_kernel(const float* __restrict__ Wz, const float* __restrict__ bz,
                            const float* __restrict__ Wh, const float* __restrict__ bh,
                            const float* __restrict__ Lz, const float* __restrict__ lzb,
                            const float* __restrict__ Lh, const float* __restrict__ lhb,
                            const float* __restrict__ att, float* __restrict__ M)
{
    const int o = threadIdx.x;
    if (o < 32) {
        float mz0 = 0.f, mz1 = 0.f, mh0 = 0.f, mh1 = 0.f;
        float cz = lzb[o], ch = lhb[o];
        for (int k = 0; k < 32; ++k) {
            const float lz = Lz[k * 32 + o];
            const float lh = Lh[k * 32 + o];
            mz0 += Wz[k]      * lz;
            mz1 += Wz[32 + k] * lz;
            mh0 += Wh[k]      * lh;
            mh1 += Wh[32 + k] * lh;
            cz  += bz[k] * lz;
            ch  += bh[k] * lh;
        }
        M[o]       = mz0;  M[32 + o]  = mz1;  M[64 + o]  = cz;
        M[96 + o]  = mh0;  M[128 + o] = mh1;  M[160 + o] = ch;
    }
    if (o == 0) {
        float mx = att[0];
        for (int t = 1; t < PERIODS; ++t) mx = fmaxf(mx, att[t]);
        float e[PERIODS], s = 0.f;
        for (int t = 0; t < PERIODS; ++t) { e[t] = __expf(att[t] - mx); s += e[t]; }
        const float inv = 1.f / s;
        for (int t = 0; t < PERIODS; ++t) M[192 + t] = e[t] * inv;
    }
}

// deg[i] = 1.0 (self loop weight)
__global__ void deg_init_kernel(float* __restrict__ deg)
{
    int i = blockIdx.x * blockDim.x + threadIdx.x;
    if (i < N_NODES) deg[i] = 1.0f;
}

// deg[col[e]] += w[e]
__global__ void deg_edge_kernel(const int* __restrict__ eidx, const float* __restrict__ ew,
                                float* __restrict__ deg, int E)
{
    int e = blockIdx.x * blockDim.x + threadIdx.x;
    if (e < E) atomicAdd(&deg[eidx[E + e]], ew[e]);
}

__global__ void dinv_kernel(const float* __restrict__ deg, float* __restrict__ dinv)
{
    int i = blockIdx.x * blockDim.x + threadIdx.x;
    if (i < N_NODES) {
        float d = deg[i];
        dinv[i] = (d > 0.f) ? __frsqrt_rn(d) : 0.f;
    }
}

// Self-loop term: Y[b,n,f,t] = dinv[n]^2 * X[b,n,f,t]  (also initializes Y)
__global__ void y_init_kernel(const float* __restrict__ X, const float* __restrict__ dinv,
                              float* __restrict__ Y)
{
    int i = blockIdx.x * blockDim.x + threadIdx.x;
    const int total = BATCH * N_NODES * FT;
    if (i < total) {
        int n = (i / FT) % N_NODES;
        float di = dinv[n];
        Y[i] = di * di * X[i];
    }
}

// Edge scatter: Y[b,col,:] += (dinv[row]*w*dinv[col]) * X[b,row,:]   (24 floats x 4 batches)
__global__ void edge_agg_kernel(const float* __restrict__ X, const int* __restrict__ eidx,
                                const float* __restrict__ ew, const float* __restrict__ dinv,
                                float* __restrict__ Y, int E)
{
    int e = blockIdx.x * blockDim.x + threadIdx.x;
    if (e >= E) return;
    const int row = eidx[e];
    const int col = eidx[E + e];
    const float nrm = dinv[row] * ew[e] * dinv[col];
    #pragma unroll
    for (int b = 0; b < BATCH; ++b) {
        const float4* __restrict__ xv =
            reinterpret_cast<const float4*>(X + ((size_t)b * N_NODES + row) * FT);
        float* __restrict__ dst = Y + ((size_t)b * N_NODES + col) * FT;
        #pragma unroll
        for (int q = 0; q < 6; ++q) {
            float4 v = xv[q];
            atomicAdd(dst + q * 4 + 0, nrm * v.x);
            atomicAdd(dst + q * 4 + 1, nrm * v.y);
            atomicAdd(dst + q * 4 + 2, nrm * v.z);
            atomicAdd(dst + q * 4 + 3, nrm * v.w);
        }
    }
}

// Branch-free transcendentals on the hardware TRANS pipe (v_exp_f32 / v_rcp_f32).
// sigmoid(x) = 1/(1 + 2^(-x*log2e));  tanh(x) = 1 - 2/(1 + 2^(x*2*log2e)).
// Both saturate correctly via rcp(inf)=0; no branches, no NaNs.
#define LOG2E 1.44269504088896f
__device__ __forceinline__ float fast_sigmoid(float x) {
    return __builtin_amdgcn_rcpf(1.f + __builtin_amdgcn_exp2f(-x * LOG2E));
}
__device__ __forceinline__ float fast_tanh(float x) {
    return 1.f - 2.f * __builtin_amdgcn_rcpf(1.f + __builtin_amdgcn_exp2f(x * (2.f * LOG2E)));
}

// ---------------------------------------------------------------------------
// Gate kernel: one wave owns 16 nodes of one batch.
// Per period t: D = A(16x4) x B(4x16) via V_WMMA_F32_16X16X4_F32.
//   A: lanes 0-15 hold {y0,y1} (K=0,1) for node m=lane; lanes 16-31 (K=2,3) = 0
//      (zeroed by lane mask multiply -> unconditional loads, no exec juggling).
//   B: lanes 0-15 VGPR0/1 = M_g rows 0/1 at column lane; lanes 16-31 = 0 pad.
//   D: VGPR v -> node row v + 8*(lane>=16), column = lane&15.
// out[b,n,o] = sum_t p_t * (1 - sigmoid(dz+cz)) * tanh(dh+ch)
// ---------------------------------------------------------------------------
__global__ void __launch_bounds__(256)
gate_wmma_kernel(const float* __restrict__ Y, const float* __restrict__ M,
                 float* __restrict__ out)
{
    const int lane = threadIdx.x & 31;
    const int wave = (blockIdx.x * blockDim.x + threadIdx.x) >> 5;
    const int tilesPerB = N_NODES / 16;            // 3125
    if (wave >= BATCH * tilesPerB) return;         // wave-uniform exit; EXEC stays full
    const int b = wave / tilesPerB;
    const int node_base = (wave % tilesPerB) * 16;
    const int nl = lane & 15;
    const float msk = (lane < 16) ? 1.f : 0.f;     // K=2,3 zero padding via multiply

    // B operands (constant across t): two gates x two 16-channel halves
    v2f bz0, bz1, bh0, bh1;
    bz0.x = M[nl]       * msk;  bz0.y = M[32 + nl]  * msk;
    bz1.x = M[16 + nl]  * msk;  bz1.y = M[48 + nl]  * msk;
    bh0.x = M[96 + nl]  * msk;  bh0.y = M[128 + nl] * msk;
    bh1.x = M[112 + nl] * msk;  bh1.y = M[144 + nl] * msk;
    const float czv0 = M[64 + nl],  czv1 = M[80 + nl];
    const float chv0 = M[160 + nl], chv1 = M[176 + nl];

    float acc0[8], acc1[8];
    #pragma unroll
    for (int v = 0; v < 8; ++v) { acc0[v] = 0.f; acc1[v] = 0.f; }

    // All lanes read a valid node (lanes 16-31 mirror lanes 0-15, masked to 0).
    const float* __restrict__ yp = Y + ((size_t)b * N_NODES + node_base + nl) * FT;

    #pragma unroll
    for (int t = 0; t < PERIODS; ++t) {
        const float p = M[192 + t];
        v2f a;
        a.x = yp[t]           * msk;   // feature 0
        a.y = yp[PERIODS + t] * msk;   // feature 1
        v8f c = {};
        v8f dz0 = __builtin_amdgcn_wmma_f32_16x16x4_f32(false, a, false, bz0, (short)0, c, false, false);
        v8f dz1 = __builtin_amdgcn_wmma_f32_16x16x4_f32(false, a, false, bz1, (short)0, c, false, false);
        v8f dh0 = __builtin_amdgcn_wmma_f32_16x16x4_f32(false, a, false, bh0, (short)0, c, false, false);
        v8f dh1 = __builtin_amdgcn_wmma_f32_16x16x4_f32(false, a, false, bh1, (short)0, c, false, false);
        #pragma unroll
        for (int v = 0; v < 8; ++v) {
            float z0 = fast_sigmoid(dz0[v] + czv0);
            float z1 = fast_sigmoid(dz1[v] + czv1);
            float h0 = fast_tanh(dh0[v] + chv0);
            float h1 = fast_tanh(dh1[v] + chv1);
            acc0[v] += p * (1.f - z0) * h0;
            acc1[v] += p * (1.f - z1) * h1;
        }
    }

    const int mrow = (lane >> 4) * 8;
    #pragma unroll
    for (int v = 0; v < 8; ++v) {
        const size_t base = ((size_t)b * N_NODES + node_base + mrow + v) * 32;
        out[base + nl]      = acc0[v];
        out[base + 16 + nl] = acc1[v];
    }
}

extern "C" void kernel_launch(void* const* d_in, const int* in_sizes, int n_in,
                              void* d_out, int out_size, void* d_ws, size_t ws_size,
                              hipStream_t stream) {
    const float* X    = (const float*)d_in[0];
    const int*   eidx = (const int*)  d_in[1];   // [2, E] -> row = eidx[e], col = eidx[E+e]
    const float* ew   = (const float*)d_in[2];
    const float* Wz   = (const float*)d_in[3];
    const float* bz   = (const float*)d_in[4];
    // d_in[5]/[6] = W_r, b_r : dead (H0 == 0 -> R unused)
    const float* Wh   = (const float*)d_in[7];
    const float* bh   = (const float*)d_in[8];
    const float* Lz   = (const float*)d_in[9];
    const float* lzb  = (const float*)d_in[10];
    // d_in[11]/[12] = lin_r_* : dead
    const float* Lh   = (const float*)d_in[13];
    const float* lhb  = (const float*)d_in[14];
    const float* att  = (const float*)d_in[15];
    const int E = in_sizes[1] / 2;

    float* ws   = (float*)d_ws;
    float* M    = ws;                         // 256 floats
    float* deg  = ws + 256;                   // N floats
    float* dinv = deg + N_NODES;              // N floats
    float* Yb   = dinv + N_NODES;             // BATCH*N*FT floats (~19.2 MB)

    prep_kernel<<<1, 32, 0, stream>>>(Wz, bz, Wh, bh, Lz, lzb, Lh, lhb, att, M);

    deg_init_kernel<<<(N_NODES + 255) / 256, 256, 0, stream>>>(deg);
    deg_edge_kernel<<<(E + 255) / 256, 256, 0, stream>>>(eidx, ew, deg, E);
    dinv_kernel<<<(N_NODES + 255) / 256, 256, 0, stream>>>(deg, dinv);

    const int ytot = BATCH * N_NODES * FT;
    y_init_kernel<<<(ytot + 255) / 256, 256, 0, stream>>>(X, dinv, Yb);
    edge_agg_kernel<<<(E + 255) / 256, 256, 0, stream>>>(X, eidx, ew, dinv, Yb, E);

    const int waves = BATCH * (N_NODES / 16);               // 12500 waves
    gate_wmma_kernel<<<(waves + 7) / 8, 256, 0, stream>>>(Yb, M, (float*)d_out);
}